// LSTM_25632364822579
// MI455X (gfx1250) — compile-verified
//
#include <hip/hip_runtime.h>

typedef __attribute__((ext_vector_type(2))) float v2f;
typedef __attribute__((ext_vector_type(8))) float v8f;

#define LSTM_B 64
#define LSTM_S 2048
#define LSTM_I 512
#define LSTM_H 512
#define LSTM_BS (LSTM_B * LSTM_S)   // 131072 rows in the input-projection GEMM

__device__ __forceinline__ float sigm_f(float x) {
  return 1.0f / (1.0f + __expf(-x));
}
__device__ __forceinline__ float tanh_f(float x) {
  // tanh(x) = 2*sigmoid(2x) - 1 ; saturates correctly for large |x|
  return 2.0f / (1.0f + __expf(-2.0f * x)) - 1.0f;
}

// ---------------------------------------------------------------------------
// Pre-pass: transpose 512x512 weight matrices W[k][n] into gate-interleaved,
// k-contiguous Wt[n][g][k] so WMMA B fragments become single b64 loads.
// blockIdx.z 0..3 -> input weights (i,f,g,o) -> wt_in ; 4..7 -> hidden -> wt_h.
// 32x33 LDS tile (padded, bank-conflict free).
// ---------------------------------------------------------------------------
__global__ __launch_bounds__(256)
void pack_transpose_weights(const float* __restrict__ s0, const float* __restrict__ s1,
                            const float* __restrict__ s2, const float* __restrict__ s3,
                            const float* __restrict__ s4, const float* __restrict__ s5,
                            const float* __restrict__ s6, const float* __restrict__ s7,
                            float* __restrict__ wt_in, float* __restrict__ wt_h)
{
  const float* src;
  switch (blockIdx.z) {
    case 0: src = s0; break; case 1: src = s1; break;
    case 2: src = s2; break; case 3: src = s3; break;
    case 4: src = s4; break; case 5: src = s5; break;
    case 6: src = s6; break; default: src = s7; break;
  }
  float* dst = (blockIdx.z < 4) ? wt_in : wt_h;
  const int g = blockIdx.z & 3;

  __shared__ float tile[32][33];
  const int tx = threadIdx.x & 31;
  const int ty = threadIdx.x >> 5;       // 0..7
  const int x0 = blockIdx.x * 32;        // n (src column)
  const int y0 = blockIdx.y * 32;        // k (src row)

  #pragma unroll
  for (int i = ty; i < 32; i += 8)
    tile[i][tx] = src[(size_t)(y0 + i) * 512 + (x0 + tx)];
  __syncthreads();
  #pragma unroll
  for (int i = ty; i < 32; i += 8) {
    const int n = x0 + i, k = y0 + tx;
    dst[((size_t)n * 4 + g) * 512 + k] = tile[tx][i];   // k-contiguous, coalesced
  }
}

// ---------------------------------------------------------------------------
// Phase 1: gates[g][b*S+s][h] = x[b,s,:] . W_in[g] + b_i[g] + b_h[g]
// Block = 256 threads = 8 waves; wave tile = 16M x 64N (4 fp32 WMMA accums),
// block tile = 64M x 128N. Full fp32 via V_WMMA_F32_16X16X4_F32.
// B fragments come from the transposed/interleaved wt_in: one b64 per tile.
// ---------------------------------------------------------------------------
__global__ __launch_bounds__(256)
void lstm_gates_gemm(const float* __restrict__ x,
                     const float* __restrict__ wt_in,
                     const float* __restrict__ bii, const float* __restrict__ bhi,
                     const float* __restrict__ bif, const float* __restrict__ bhf,
                     const float* __restrict__ big, const float* __restrict__ bhg,
                     const float* __restrict__ bio, const float* __restrict__ bho,
                     float* __restrict__ gates)
{
  const int gate = blockIdx.z;
  const float* b0; const float* b1;
  switch (gate) {
    case 0:  b0 = bii; b1 = bhi; break;
    case 1:  b0 = bif; b1 = bhf; break;
    case 2:  b0 = big; b1 = bhg; break;
    default: b0 = bio; b1 = bho; break;
  }
  float* out = gates + (size_t)gate * (size_t)LSTM_BS * LSTM_H;

  const int lane = threadIdx.x & 31;   // wave32
  const int wave = threadIdx.x >> 5;
  const int lm   = lane & 15;          // A row / B col / C col within tile
  const int lh   = lane >> 4;          // lane half: K pair (A/B) or M half (C/D)
  const int k0   = lh << 1;

  const size_t m0 = (size_t)blockIdx.x * 64 + (size_t)(wave >> 1) * 16;
  const int    n0 = blockIdx.y * 128 + (wave & 1) * 64;
  const int    nc = n0 + lm;

  const float* arow = x + (m0 + lm) * (size_t)LSTM_I + k0;
  const float* bb0 = wt_in + ((size_t)(nc     ) * 4 + gate) * 512 + k0;
  const float* bb1 = wt_in + ((size_t)(nc + 16) * 4 + gate) * 512 + k0;
  const float* bb2 = wt_in + ((size_t)(nc + 32) * 4 + gate) * 512 + k0;
  const float* bb3 = wt_in + ((size_t)(nc + 48) * 4 + gate) * 512 + k0;

  v8f acc0 = {}, acc1 = {}, acc2 = {}, acc3 = {};
  #pragma unroll 4
  for (int kb = 0; kb < LSTM_I; kb += 4) {
    v2f a  = *(const v2f*)(arow + kb);
    v2f w0 = *(const v2f*)(bb0 + kb);
    v2f w1 = *(const v2f*)(bb1 + kb);
    v2f w2 = *(const v2f*)(bb2 + kb);
    v2f w3 = *(const v2f*)(bb3 + kb);
    acc0 = __builtin_amdgcn_wmma_f32_16x16x4_f32(false, a, false, w0, (short)0, acc0, false, false);
    acc1 = __builtin_amdgcn_wmma_f32_16x16x4_f32(false, a, false, w1, (short)0, acc1, false, false);
    acc2 = __builtin_amdgcn_wmma_f32_16x16x4_f32(false, a, false, w2, (short)0, acc2, false, false);
    acc3 = __builtin_amdgcn_wmma_f32_16x16x4_f32(false, a, false, w3, (short)0, acc3, false, false);
  }

  const float bs0 = b0[nc]      + b1[nc];
  const float bs1 = b0[nc + 16] + b1[nc + 16];
  const float bs2 = b0[nc + 32] + b1[nc + 32];
  const float bs3 = b0[nc + 48] + b1[nc + 48];

  #pragma unroll
  for (int v = 0; v < 8; ++v) {
    const size_t m = m0 + (size_t)(v + 8 * lh);       // C/D layout: VGPR v, lane half
    float* p = out + m * (size_t)LSTM_H;
    __builtin_nontemporal_store(acc0[v] + bs0, p + nc);
    __builtin_nontemporal_store(acc1[v] + bs1, p + nc + 16);
    __builtin_nontemporal_store(acc2[v] + bs2, p + nc + 32);
    __builtin_nontemporal_store(acc3[v] + bs3, p + nc + 48);
  }
}

// ---------------------------------------------------------------------------
// Grid-wide sense barrier (16 co-resident blocks), device-scope atomics.
// ---------------------------------------------------------------------------
__device__ __forceinline__ void grid_barrier(unsigned* bar, unsigned nblocks) {
  __threadfence();               // release this thread's h-stores device-wide
  __syncthreads();
  if (threadIdx.x == 0) {
    unsigned gen = __hip_atomic_load(bar + 1, __ATOMIC_RELAXED, __HIP_MEMORY_SCOPE_AGENT);
    unsigned arrived = __hip_atomic_fetch_add(bar, 1u, __ATOMIC_ACQ_REL, __HIP_MEMORY_SCOPE_AGENT);
    if (arrived == nblocks - 1u) {
      __hip_atomic_store(bar, 0u, __ATOMIC_RELAXED, __HIP_MEMORY_SCOPE_AGENT);
      __hip_atomic_store(bar + 1, gen + 1u, __ATOMIC_RELEASE, __HIP_MEMORY_SCOPE_AGENT);
    } else {
      while (__hip_atomic_load(bar + 1, __ATOMIC_ACQUIRE, __HIP_MEMORY_SCOPE_AGENT) == gen) {
        __builtin_amdgcn_s_sleep(1);
      }
    }
  }
  __syncthreads();
}

// ---------------------------------------------------------------------------
// Phase 2: persistent recurrent scan. 16 blocks x 8 waves = 128 waves; each
// wave owns one (16-batch x 16-hidden) tile for ALL 4 gates across all 2048
// steps, so cell state c stays in 8 VGPRs. h double-buffered in global memory;
// the 4 MB of recurrent weights are L2-resident for the whole scan, and the
// four gate B-fragments share one base register (imm offsets 0/2/4/6 KB).
// ---------------------------------------------------------------------------
__global__ __launch_bounds__(256)
void lstm_recurrent(const float* __restrict__ gates,
                    const float* __restrict__ wt_h,
                    float* __restrict__ hbuf0, float* __restrict__ hbuf1,
                    unsigned* __restrict__ bar,
                    float* __restrict__ out)
{
  const int lane = threadIdx.x & 31;
  const int wave = threadIdx.x >> 5;
  const int gw   = blockIdx.x * 8 + wave;   // 0..127 wave slots
  const int mt   = gw >> 5;                 // 0..3  batch tile (16 rows)
  const int nt   = gw & 31;                 // 0..31 hidden tile (16 cols)
  const int lm   = lane & 15;
  const int lh   = lane >> 4;
  const int k0   = lh << 1;
  const int nc   = nt * 16 + lm;

  const size_t plane = (size_t)LSTM_BS * LSTM_H;
  const float* gi = gates;
  const float* gf = gates + plane;
  const float* gg = gates + 2 * plane;
  const float* go = gates + 3 * plane;

  // wt_h layout: [n][gate][k] ; single base, gates at +512/+1024/+1536 floats
  const float* wb = wt_h + (size_t)nc * 4 * 512 + k0;

  float creg[8];
  #pragma unroll
  for (int v = 0; v < 8; ++v) creg[v] = 0.0f;

  for (int s = 0; s < LSTM_S; ++s) {
    const float* hc = (s & 1) ? hbuf1 : hbuf0;
    float*       hn = (s & 1) ? hbuf0 : hbuf1;

    v8f ai = {}, af = {}, ag = {}, ao = {};
    const float* ar = hc + (size_t)(mt * 16 + lm) * LSTM_H + k0;
    #pragma unroll 4
    for (int kb = 0; kb < LSTM_H; kb += 4) {
      v2f a   = *(const v2f*)(ar + kb);              // shared h fragment (b64)
      v2f bi_ = *(const v2f*)(wb + kb);              // gate i  (b64)
      v2f bf_ = *(const v2f*)(wb + 512 + kb);        // gate f
      v2f bg_ = *(const v2f*)(wb + 1024 + kb);       // gate g
      v2f bo_ = *(const v2f*)(wb + 1536 + kb);       // gate o
      ai = __builtin_amdgcn_wmma_f32_16x16x4_f32(false, a, false, bi_, (short)0, ai, false, false);
      af = __builtin_amdgcn_wmma_f32_16x16x4_f32(false, a, false, bf_, (short)0, af, false, false);
      ag = __builtin_amdgcn_wmma_f32_16x16x4_f32(false, a, false, bg_, (short)0, ag, false, false);
      ao = __builtin_amdgcn_wmma_f32_16x16x4_f32(false, a, false, bo_, (short)0, ao, false, false);
    }

    // Elementwise LSTM cell: the 4 gate accumulators hold the SAME (b,h)
    // element in matching lane/VGPR slots (C/D layout), so this is in-register.
    #pragma unroll
    for (int v = 0; v < 8; ++v) {
      const int bb = mt * 16 + v + 8 * lh;
      const size_t gidx = ((size_t)bb * LSTM_S + s) * LSTM_H + nc;
      const float pi = ai[v] + __builtin_nontemporal_load(gi + gidx);
      const float pf = af[v] + __builtin_nontemporal_load(gf + gidx);
      const float pg = ag[v] + __builtin_nontemporal_load(gg + gidx);
      const float po = ao[v] + __builtin_nontemporal_load(go + gidx);
      const float it = sigm_f(pi);
      const float ft = sigm_f(pf);
      const float gt = tanh_f(pg);
      const float ot = sigm_f(po);
      const float cn = ft * creg[v] + it * gt;
      const float hv = ot * tanh_f(cn);
      creg[v] = cn;
      hn[(size_t)bb * LSTM_H + nc] = hv;
      if (s == LSTM_S - 1) {
        out[(size_t)bb * LSTM_H + nc] = hv;                              // h_t
        out[(size_t)(LSTM_B * LSTM_H) + (size_t)bb * LSTM_H + nc] = cn;  // c_t
      }
    }
    grid_barrier(bar, gridDim.x);
  }
}

// ---------------------------------------------------------------------------
extern "C" void kernel_launch(void* const* d_in, const int* in_sizes, int n_in,
                              void* d_out, int out_size, void* d_ws, size_t ws_size,
                              hipStream_t stream) {
  (void)in_sizes; (void)n_in; (void)out_size; (void)ws_size;
  const float* x    = (const float*)d_in[0];
  const float* w_ii = (const float*)d_in[1];
  const float* w_hi = (const float*)d_in[2];
  const float* w_if = (const float*)d_in[3];
  const float* w_hf = (const float*)d_in[4];
  const float* w_ig = (const float*)d_in[5];
  const float* w_hg = (const float*)d_in[6];
  const float* w_io = (const float*)d_in[7];
  const float* w_ho = (const float*)d_in[8];
  const float* b_ii = (const float*)d_in[9];
  const float* b_hi = (const float*)d_in[10];
  const float* b_if = (const float*)d_in[11];
  const float* b_hf = (const float*)d_in[12];
  const float* b_ig = (const float*)d_in[13];
  const float* b_hg = (const float*)d_in[14];
  const float* b_io = (const float*)d_in[15];
  const float* b_ho = (const float*)d_in[16];

  // Workspace layout: 4 gate planes (fp32) | h double-buffer | barrier (256B pad)
  //                   | wt_in[512][4][512] | wt_h[512][4][512]
  float* gates = (float*)d_ws;
  const size_t plane = (size_t)LSTM_BS * LSTM_H;            // 64M elems per gate
  float* hbuf0 = gates + 4 * plane;
  float* hbuf1 = hbuf0 + (size_t)LSTM_B * LSTM_H;
  unsigned* bar = (unsigned*)(hbuf1 + (size_t)LSTM_B * LSTM_H);
  float* wt_in = (float*)(bar + 64);
  float* wt_h  = wt_in + (size_t)4 * 512 * 512;

  hipMemsetAsync(hbuf0, 0, (size_t)LSTM_B * LSTM_H * sizeof(float), stream); // h0 = 0
  hipMemsetAsync(bar, 0, 2 * sizeof(unsigned), stream);                      // barrier state

  // One-time weight transpose/interleave (8 MB total).
  pack_transpose_weights<<<dim3(16, 16, 8), dim3(256), 0, stream>>>(
      w_ii, w_if, w_ig, w_io, w_hi, w_hf, w_hg, w_ho, wt_in, wt_h);

  dim3 g1(LSTM_BS / 64, LSTM_H / 128, 4), b1(256);
  lstm_gates_gemm<<<g1, b1, 0, stream>>>(x, wt_in,
      b_ii, b_hi, b_if, b_hf, b_ig, b_hg, b_io, b_ho,
      gates);

  lstm_recurrent<<<dim3(16), dim3(256), 0, stream>>>(gates, wt_h,
      hbuf0, hbuf1, bar, (float*)d_out);
}